// SetOfSetProjectionFeatureUpdate_33088428049082
// MI455X (gfx1250) — compile-verified
//
#include <hip/hip_runtime.h>

typedef float v2f __attribute__((ext_vector_type(2)));
typedef float v8f __attribute__((ext_vector_type(8)));

#define DIM 128            // D_IN == D_OUT == 128
#define LDS_STRIDE 132     // 128 + 4 pad -> bank = (4*o + d) % 64, conflict-free
#define WAVES_PER_BLOCK 4

// One wave computes a 16-row x 128-col tile of:
//   out = 0.25 * (values @ W^T + b + sp[col_idx] + vf[row_idx] + gf)
__global__ __launch_bounds__(128, 1)
void sos_proj_wmma_kernel(const float* __restrict__ values,
                          const int*   __restrict__ row_idx,
                          const int*   __restrict__ col_idx,
                          const float* __restrict__ sp,
                          const float* __restrict__ vf,
                          const float* __restrict__ gf,
                          const float* __restrict__ W,
                          const float* __restrict__ bias,
                          float*       __restrict__ out,
                          int ntiles)
{
    __shared__ float ldsW[DIM * LDS_STRIDE];

    const int tid = threadIdx.x;

    // ---- stage W (64 KB, L2-resident) into LDS with padded stride ----
    for (int i = tid; i < DIM * DIM; i += blockDim.x) {
        const int o = i >> 7;
        const int d = i & 127;
        ldsW[o * LDS_STRIDE + d] = W[i];
    }
    __syncthreads();

    const int lane = tid & 31;
    const int wave = tid >> 5;
    const int lo   = lane & 15;   // row within A tile / N within B,C,D tiles
    const int hi   = lane >> 4;   // K-half selector for A/B fragments

    // per-lane invariant across all tiles: bias[o] + global_features[o]
    float bg[8];
#pragma unroll
    for (int j = 0; j < 8; ++j) {
        const int o = 16 * j + lo;
        bg[j] = bias[o] + gf[o];
    }

    for (int t = blockIdx.x * WAVES_PER_BLOCK + wave; t < ntiles;
         t += gridDim.x * WAVES_PER_BLOCK) {
        const int base = t * 16;

        // ---- load A tile (16 rows x 128 K of `values`) into registers ----
        // A frag layout (16x4 f32): lanes 0-15 hold K=4c,4c+1; lanes 16-31 hold K=4c+2,4c+3
        v2f a[32];
        const float* aRow = values + (size_t)(base + lo) * DIM + 2 * hi;
#pragma unroll
        for (int c = 0; c < 32; ++c) {
            a[c] = __builtin_nontemporal_load((const v2f*)(aRow + 4 * c));
        }

        v8f acc[8];
#pragma unroll
        for (int j = 0; j < 8; ++j) acc[j] = (v8f)0.0f;

        // ---- 8 col tiles x 32 K-chunks of V_WMMA_F32_16X16X4_F32 ----
#pragma unroll
        for (int j = 0; j < 8; ++j) {
            const float* bBase = ldsW + (16 * j + lo) * LDS_STRIDE + 2 * hi;
#pragma unroll
            for (int c = 0; c < 32; ++c) {
                // B[d, o] = W[o, d]; frag mirrors A's K split, N = lane%16
                v2f bfrag = *(const v2f*)(bBase + 4 * c);
                acc[j] = __builtin_amdgcn_wmma_f32_16x16x4_f32(
                    false, a[c], false, bfrag, (short)0, acc[j], false, false);
            }
        }

        // ---- epilogue: gathers (L2-resident tables) + scale + NT store ----
        // C/D layout: VGPR r, lanes 0-15 -> row r, lanes 16-31 -> row r+8; col = lane%16
#pragma unroll
        for (int r = 0; r < 8; ++r) {
            const int m  = r + 8 * hi;
            const int n  = base + m;
            const int ri = row_idx[n];
            const int ci = col_idx[n];
            const float* spRow  = sp  + (size_t)ci * DIM;
            const float* vfRow  = vf  + (size_t)ri * DIM;
            float*       outRow = out + (size_t)n  * DIM;
#pragma unroll
            for (int j = 0; j < 8; ++j) {
                const int o = 16 * j + lo;
                const float v = 0.25f * (acc[j][r] + bg[j] + spRow[o] + vfRow[o]);
                __builtin_nontemporal_store(v, outRow + o);
            }
        }
    }
}

extern "C" void kernel_launch(void* const* d_in, const int* in_sizes, int n_in,
                              void* d_out, int out_size, void* d_ws, size_t ws_size,
                              hipStream_t stream) {
    const float* values  = (const float*)d_in[0];
    const int*   row_idx = (const int*)  d_in[1];
    const int*   col_idx = (const int*)  d_in[2];
    const float* sp      = (const float*)d_in[3];
    const float* vf      = (const float*)d_in[4];
    const float* gf      = (const float*)d_in[5];
    const float* W       = (const float*)d_in[6];
    const float* bias    = (const float*)d_in[7];
    float* out = (float*)d_out;

    const int nnz    = in_sizes[1];      // row_idx count == NNZ (2,000,000)
    const int ntiles = nnz / 16;         // 125,000 tiles of 16 rows

    int grid = 4096;
    if (grid * WAVES_PER_BLOCK > ntiles) grid = (ntiles + WAVES_PER_BLOCK - 1) / WAVES_PER_BLOCK;
    if (grid < 1) grid = 1;

    hipLaunchKernelGGL(sos_proj_wmma_kernel, dim3(grid), dim3(128), 0, stream,
                       values, row_idx, col_idx, sp, vf, gf, W, bias, out, ntiles);
}